// NLayerGCN_2035814498364
// MI455X (gfx1250) — compile-verified
//
#include <hip/hip_runtime.h>
#include <math.h>

typedef __attribute__((ext_vector_type(2))) float v2f;
typedef __attribute__((ext_vector_type(8))) float v8f;

#define N_NODES 100000
#define N_EDGES 1600000
#define FDIM    128
#define CDIM    20
#define CPAD    32
#define NEG_SLOPE 0.01f

// ---------------- small helpers ----------------

__global__ void pad_w2_kernel(const float* __restrict__ W2, float* __restrict__ W2P) {
    int t = blockIdx.x * blockDim.x + threadIdx.x;
    if (t >= FDIM * CPAD) return;
    int k = t >> 5, c = t & 31;
    W2P[t] = (c < CDIM) ? W2[k * CDIM + c] : 0.0f;
}

__global__ void init_deg_kernel(float* __restrict__ deg, int n) {
    int i = blockIdx.x * blockDim.x + threadIdx.x;
    if (i < n) deg[i] = 1.0f;   // self-loop weight
}

__global__ void deg_accum_kernel(const int* __restrict__ dst, const float* __restrict__ ew,
                                 float* __restrict__ deg, int e) {
    int i = blockIdx.x * blockDim.x + threadIdx.x;
    if (i < e) atomicAdd(&deg[dst[i]], ew[i]);
}

__global__ void dinv_kernel(const float* __restrict__ deg, float* __restrict__ dinv, int n) {
    int i = blockIdx.x * blockDim.x + threadIdx.x;
    if (i < n) dinv[i] = rsqrtf(deg[i]);    // deg >= 1 always
}

__global__ void gather_kernel(const int* __restrict__ ids, const float* __restrict__ emb,
                              float* __restrict__ X, int n) {
    int t = blockIdx.x * blockDim.x + threadIdx.x;   // n*32 threads, float4 granule
    if (t >= n * (FDIM / 4)) return;
    int node = t >> 5, g = t & 31;
    ((float4*)X)[t] = ((const float4*)(emb + (size_t)ids[node] * FDIM))[g];
}

__global__ void zero_f4_kernel(float4* __restrict__ p, int n4) {
    int i = blockIdx.x * blockDim.x + threadIdx.x;
    if (i < n4) p[i] = make_float4(0.f, 0.f, 0.f, 0.f);
}

// ---------------- fp32 WMMA GEMM: H[nrows x NCOLS] = X[nrows x 128] * W[128 x NCOLS] ----------------
// One wave computes a 16-row strip, holding all NCOLS/16 accumulator tiles in VGPRs.
// W is staged in LDS K-pair interleaved: Wlds[(k/2)*NCOLS + col] = {W[k][col], W[k+1][col]},
// so each B operand is one aligned ds_load_b64 landing directly in an even VGPR pair.

template <int NCOLS>
__global__ __launch_bounds__(256) void gemm_wmma_kernel(const float* __restrict__ X,
                                                        const float* __restrict__ Wg,
                                                        float* __restrict__ H, int nrows) {
    __shared__ float2 Wlds[FDIM / 2 * NCOLS];

    // stage + interleave: p enumerates (kpair, col); global reads coalesced over col
    for (int p = threadIdx.x; p < FDIM / 2 * NCOLS; p += 256) {
        int kp  = p / NCOLS;
        int col = p - kp * NCOLS;
        float e0 = Wg[(2 * kp)     * NCOLS + col];
        float e1 = Wg[(2 * kp + 1) * NCOLS + col];
        Wlds[p] = make_float2(e0, e1);
    }
    __syncthreads();

    const int wave    = threadIdx.x >> 5;
    const int lane    = threadIdx.x & 31;
    const int rowBase = (blockIdx.x * 8 + wave) * 16;
    if (rowBase + 16 > nrows) return;          // wave-uniform: EXEC stays all-ones for live waves

    const int hi = lane >> 4;                  // selects K sub-pair {0,1} vs {2,3}
    const int lo = lane & 15;                  // row (for A) / col (for B,D) within tile
    constexpr int NT = NCOLS / 16;

    v8f acc[NT] = {};
    const float* __restrict__ xrow = X + (size_t)(rowBase + lo) * FDIM + 2 * hi;
    const v2f* __restrict__ Wp = (const v2f*)Wlds;

#pragma unroll 4
    for (int k0 = 0; k0 < FDIM; k0 += 4) {
        v2f a;
        a.x = xrow[k0];
        a.y = xrow[k0 + 1];
        const v2f* wrow = Wp + (size_t)(k0 / 2 + hi) * NCOLS + lo;
#pragma unroll
        for (int t = 0; t < NT; ++t) {
            v2f b = wrow[t * 16];
            acc[t] = __builtin_amdgcn_wmma_f32_16x16x4_f32(
                false, a, false, b, (short)0, acc[t], false, false);
        }
    }

#pragma unroll
    for (int t = 0; t < NT; ++t)
#pragma unroll
        for (int r = 0; r < 8; ++r)
            H[(size_t)(rowBase + 8 * hi + r) * NCOLS + t * 16 + lo] = acc[t][r];
}

// ---------------- edge scatter: AGG[dst] += coef * H[src] ----------------
// One wave per edge; each lane moves a float4 (F=128) or one float (CPAD=32).

__global__ __launch_bounds__(256) void scatter128_kernel(const int* __restrict__ src,
                                                         const int* __restrict__ dst,
                                                         const float* __restrict__ ew,
                                                         const float* __restrict__ dinv,
                                                         const float* __restrict__ H,
                                                         float* __restrict__ AGG, int nedges) {
    int e = blockIdx.x * 8 + (threadIdx.x >> 5);
    if (e >= nedges) return;
    int lane = threadIdx.x & 31;
    int s = src[e], d = dst[e];
    float coef = dinv[s] * ew[e] * dinv[d];
    float4 h4 = ((const float4*)(H + (size_t)s * FDIM))[lane];
    float* o = AGG + (size_t)d * FDIM + lane * 4;
    atomicAdd(o + 0, h4.x * coef);
    atomicAdd(o + 1, h4.y * coef);
    atomicAdd(o + 2, h4.z * coef);
    atomicAdd(o + 3, h4.w * coef);
}

__global__ __launch_bounds__(256) void scatter32_kernel(const int* __restrict__ src,
                                                        const int* __restrict__ dst,
                                                        const float* __restrict__ ew,
                                                        const float* __restrict__ dinv,
                                                        const float* __restrict__ H,
                                                        float* __restrict__ AGG, int nedges) {
    int e = blockIdx.x * 8 + (threadIdx.x >> 5);
    if (e >= nedges) return;
    int lane = threadIdx.x & 31;          // cols 20..31 carry zeros (padded), harmless
    int s = src[e], d = dst[e];
    float coef = dinv[s] * ew[e] * dinv[d];
    float hv = H[(size_t)s * CPAD + lane];
    atomicAdd(&AGG[(size_t)d * CPAD + lane], hv * coef);
}

// ---------------- combine: X = leaky(AGG + H*dinv^2 + b) ----------------

__global__ void combine128_kernel(const float* __restrict__ AGG, const float* __restrict__ H,
                                  const float* __restrict__ dinv, const float* __restrict__ b,
                                  float* __restrict__ Xout, int n) {
    int t = blockIdx.x * blockDim.x + threadIdx.x;
    if (t >= n * (FDIM / 4)) return;
    int node = t >> 5, g = t & 31;
    float di = dinv[node];
    float s = di * di;
    float4 a4 = ((const float4*)AGG)[t];
    float4 h4 = ((const float4*)H)[t];
    float4 b4 = ((const float4*)b)[g];
    float4 r;
    r.x = a4.x + h4.x * s + b4.x;
    r.y = a4.y + h4.y * s + b4.y;
    r.z = a4.z + h4.z * s + b4.z;
    r.w = a4.w + h4.w * s + b4.w;
    r.x = (r.x >= 0.f) ? r.x : NEG_SLOPE * r.x;
    r.y = (r.y >= 0.f) ? r.y : NEG_SLOPE * r.y;
    r.z = (r.z >= 0.f) ? r.z : NEG_SLOPE * r.z;
    r.w = (r.w >= 0.f) ? r.w : NEG_SLOPE * r.w;
    ((float4*)Xout)[t] = r;
}

// ---------------- final layer: combine + log_softmax over C=20 ----------------

__global__ void finalize_kernel(const float* __restrict__ AGG, const float* __restrict__ H,
                                const float* __restrict__ dinv, const float* __restrict__ b,
                                float* __restrict__ out, int n) {
    int node = blockIdx.x * blockDim.x + threadIdx.x;
    if (node >= n) return;
    float di = dinv[node];
    float s = di * di;
    const float* ar = AGG + (size_t)node * CPAD;
    const float* hr = H + (size_t)node * CPAD;
    float v[CDIM];
    float mx = -3.402823466e38f;
#pragma unroll
    for (int c = 0; c < CDIM; ++c) {
        float x = ar[c] + hr[c] * s + b[c];
        v[c] = x;
        mx = fmaxf(mx, x);
    }
    float sum = 0.f;
#pragma unroll
    for (int c = 0; c < CDIM; ++c) sum += expf(v[c] - mx);
    float lse = mx + logf(sum);
    float* o = out + (size_t)node * CDIM;
#pragma unroll
    for (int c = 0; c < CDIM; ++c) o[c] = v[c] - lse;
}

// ---------------- launch ----------------

static inline size_t align256(size_t x) { return (x + 255) & ~(size_t)255; }

extern "C" void kernel_launch(void* const* d_in, const int* in_sizes, int n_in,
                              void* d_out, int out_size, void* d_ws, size_t ws_size,
                              hipStream_t stream) {
    const int*   node_ids = (const int*)d_in[0];
    const int*   e_src    = (const int*)d_in[1];             // edge_index row 0
    const int*   e_dst    = ((const int*)d_in[1]) + N_EDGES; // edge_index row 1
    const float* edge_w   = (const float*)d_in[2];
    const float* emb      = (const float*)d_in[3];
    const float* W0       = (const float*)d_in[4];
    const float* b0       = (const float*)d_in[5];
    const float* W1       = (const float*)d_in[6];
    const float* b1       = (const float*)d_in[7];
    const float* W2       = (const float*)d_in[8];
    const float* b2       = (const float*)d_in[9];
    float*       out      = (float*)d_out;

    char* ws = (char*)d_ws;
    size_t off = 0;
    float* X    = (float*)(ws + off); off = align256(off + (size_t)N_NODES * FDIM * 4);
    float* H    = (float*)(ws + off); off = align256(off + (size_t)N_NODES * FDIM * 4);
    float* AGG  = (float*)(ws + off); off = align256(off + (size_t)N_NODES * FDIM * 4);
    float* DEG  = (float*)(ws + off); off = align256(off + (size_t)N_NODES * 4);
    float* DINV = (float*)(ws + off); off = align256(off + (size_t)N_NODES * 4);
    float* W2P  = (float*)(ws + off); off = align256(off + (size_t)FDIM * CPAD * 4);
    (void)ws_size; (void)in_sizes; (void)n_in; (void)out_size;

    const int B = 256;
    const int gN       = (N_NODES + B - 1) / B;
    const int gE       = (N_EDGES + B - 1) / B;
    const int gNF4     = ((N_NODES * (FDIM / 4)) + B - 1) / B;  // 12500
    const int gGemm    = (N_NODES + 127) / 128;                 // 782 blocks, 8 waves each
    const int gEdge    = (N_EDGES + 7) / 8;                     // one wave per edge
    const int gZero128 = ((N_NODES * FDIM / 4) + B - 1) / B;
    const int gZero32  = ((N_NODES * CPAD / 4) + B - 1) / B;

    // graph-invariant prep (re-done every call: ws is not preserved)
    pad_w2_kernel<<<(FDIM * CPAD + B - 1) / B, B, 0, stream>>>(W2, W2P);
    init_deg_kernel<<<gN, B, 0, stream>>>(DEG, N_NODES);
    deg_accum_kernel<<<gE, B, 0, stream>>>(e_dst, edge_w, DEG, N_EDGES);
    dinv_kernel<<<gN, B, 0, stream>>>(DEG, DINV, N_NODES);
    gather_kernel<<<gNF4, B, 0, stream>>>(node_ids, emb, X, N_NODES);

    // ---- layer 0 ----
    gemm_wmma_kernel<FDIM><<<gGemm, B, 0, stream>>>(X, W0, H, N_NODES);
    zero_f4_kernel<<<gZero128, B, 0, stream>>>((float4*)AGG, N_NODES * FDIM / 4);
    scatter128_kernel<<<gEdge, B, 0, stream>>>(e_src, e_dst, edge_w, DINV, H, AGG, N_EDGES);
    combine128_kernel<<<gNF4, B, 0, stream>>>(AGG, H, DINV, b0, X, N_NODES);

    // ---- layer 1 ----
    gemm_wmma_kernel<FDIM><<<gGemm, B, 0, stream>>>(X, W1, H, N_NODES);
    zero_f4_kernel<<<gZero128, B, 0, stream>>>((float4*)AGG, N_NODES * FDIM / 4);
    scatter128_kernel<<<gEdge, B, 0, stream>>>(e_src, e_dst, edge_w, DINV, H, AGG, N_EDGES);
    combine128_kernel<<<gNF4, B, 0, stream>>>(AGG, H, DINV, b1, X, N_NODES);

    // ---- layer 2 (padded to 32 cols) + log_softmax ----
    gemm_wmma_kernel<CPAD><<<gGemm, B, 0, stream>>>(X, W2P, H, N_NODES);
    zero_f4_kernel<<<gZero32, B, 0, stream>>>((float4*)AGG, N_NODES * CPAD / 4);
    scatter32_kernel<<<gEdge, B, 0, stream>>>(e_src, e_dst, edge_w, DINV, H, AGG, N_EDGES);
    finalize_kernel<<<gN, B, 0, stream>>>(AGG, H, DINV, b2, out, N_NODES);
}